// YOLOXPostprocess_52999896432901
// MI455X (gfx1250) — compile-verified
//
#include <hip/hip_runtime.h>
#include <hip/hip_bf16.h>

// ---------------------------------------------------------------------------
// YOLOX postprocess for MI455X (gfx1250, wave32).
// Pipeline: decode+score -> per-image LDS bitonic sort -> WMMA-assisted
// NMS suppression bitmask (async-LDS staged tiles) -> 1-wave serial scan
// -> output pack.  Workspace layout (assumed ws_size >= ~40MB; mask term
// dominates at 35.3MB -- the full-IoU reference would need 282MB/image).
// ---------------------------------------------------------------------------

namespace {
constexpr int   BATCH = 4;
constexpr int   NPTS  = 8400;          // 80*80 + 40*40 + 20*20
constexpr int   NCLS  = 80;
constexpr int   RB    = NPTS / 16;     // 525 row blocks of 16 (exact)
constexpr int   NCW   = (NPTS + 31) / 32; // 263 column mask words
constexpr int   SORTN = 16384;         // next pow2 >= NPTS
constexpr float THR   = 0.65f;
constexpr float STHR  = 0.01f;
}

typedef float v2f __attribute__((ext_vector_type(2)));
typedef float v8f __attribute__((ext_vector_type(8)));

// ---------------------------------------------------------------------------
// K1: fused flatten + sigmoid + max/argmax + box decode.
// sigmoid is monotonic -> max(sigmoid(cls)) == sigmoid(max(cls)).
// Invalid (score < 0.01) gets sort key -1.0, matching the reference exactly.
// ---------------------------------------------------------------------------
__global__ void k_decode(const float* __restrict__ c0, const float* __restrict__ c1,
                         const float* __restrict__ c2, const float* __restrict__ r0,
                         const float* __restrict__ r1, const float* __restrict__ r2,
                         const float* __restrict__ o0, const float* __restrict__ o1,
                         const float* __restrict__ o2,
                         float* __restrict__ skey, int* __restrict__ lab,
                         float4* __restrict__ box) {
  int t = blockIdx.x * blockDim.x + threadIdx.x;
  if (t >= BATCH * NPTS) return;
  int b = t / NPTS, p = t % NPTS;
  const float *cls, *reg, *obj; int W, st, q;
  if (p < 6400)      { cls = c0; reg = r0; obj = o0; W = 80; st = 8;  q = p; }
  else if (p < 8000) { cls = c1; reg = r1; obj = o1; W = 40; st = 16; q = p - 6400; }
  else               { cls = c2; reg = r2; obj = o2; W = 20; st = 32; q = p - 8000; }
  int HW = W * W;
  int y = q / W, x = q % W;

  const float* cp = cls + (size_t)b * NCLS * HW + q;   // channel-strided, lane-coalesced
  float m = cp[0]; int lb = 0;
  for (int c = 1; c < NCLS; ++c) {
    float v = cp[(size_t)c * HW];
    if (v > m) { m = v; lb = c; }                      // strict > == argmax first-hit
  }
  float ob    = obj[(size_t)b * HW + q];
  float score = (1.f / (1.f + expf(-m))) * (1.f / (1.f + expf(-ob)));

  const float* rp = reg + (size_t)b * 4 * HW + q;
  float cx = rp[0] * st + x * st;
  float cy = rp[(size_t)HW] * st + y * st;
  float bw = expf(rp[(size_t)2 * HW]) * st;
  float bh = expf(rp[(size_t)3 * HW]) * st;

  skey[t] = (score >= STHR) ? score : -1.0f;
  lab[t]  = lb;
  box[t]  = make_float4(cx - 0.5f * bw, cy - 0.5f * bh, cx + 0.5f * bw, cy + 0.5f * bh);
}

// ---------------------------------------------------------------------------
// K2: per-image bitonic sort fully in LDS (128KB of the 320KB/WGP budget).
// Key = (~orderedFloatBits(score) << 32) | index  -> descending score,
// ascending index on ties == JAX stable argsort(-key).
// ---------------------------------------------------------------------------
__global__ __launch_bounds__(512) void k_sort(const float* __restrict__ skey,
                                              const int* __restrict__ lab,
                                              const float4* __restrict__ box,
                                              float* __restrict__ sskey,
                                              int* __restrict__ slab,
                                              float4* __restrict__ sbox) {
  __shared__ unsigned long long key[SORTN];   // 128 KB
  const int b = blockIdx.x, tid = threadIdx.x;
  const size_t bo = (size_t)b * NPTS;

  for (int i = tid; i < SORTN; i += 512) {
    unsigned long long kk = ~0ull;            // pad sorts to the end
    if (i < NPTS) {
      unsigned u = __float_as_uint(skey[bo + i]);
      unsigned o = u ^ ((u >> 31) ? 0xFFFFFFFFu : 0x80000000u);  // ascending-order map
      kk = ((unsigned long long)(~o) << 32) | (unsigned)i;       // invert -> descending
    }
    key[i] = kk;
  }
  for (unsigned k = 2; k <= SORTN; k <<= 1)
    for (unsigned j = k >> 1; j > 0; j >>= 1) {
      __syncthreads();
      for (unsigned i = tid; i < SORTN; i += 512) {
        unsigned ij = i ^ j;
        if (ij > i) {
          unsigned long long a = key[i], c = key[ij];
          bool asc = ((i & k) == 0);
          if ((a > c) == asc) { key[i] = c; key[ij] = a; }
        }
      }
    }
  __syncthreads();
  for (int t = tid; t < NPTS; t += 512) {
    unsigned idx = (unsigned)key[t];
    sskey[bo + t] = skey[bo + idx];
    slab[bo + t]  = lab[bo + idx];
    sbox[bo + t]  = box[bo + idx];
  }
}

// ---------------------------------------------------------------------------
// gfx1250 async global->LDS staging (GVS mode: SGPR base + 32-bit VGPR offset).
// ---------------------------------------------------------------------------
__device__ __forceinline__ void async_lds_b128(unsigned lds, unsigned off,
                                               unsigned long long base) {
  asm volatile("global_load_async_to_lds_b128 %0, %1, %2"
               :: "v"(lds), "v"(off), "s"(base) : "memory");
}
__device__ __forceinline__ void async_lds_b32(unsigned lds, unsigned off,
                                              unsigned long long base) {
  asm volatile("global_load_async_to_lds_b32 %0, %1, %2"
               :: "v"(lds), "v"(off), "s"(base) : "memory");
}
__device__ __forceinline__ void wait_asynccnt0() {
  asm volatile("s_wait_asynccnt 0x0" ::: "memory");
}

// ---------------------------------------------------------------------------
// K3: suppression bitmask.  One wave = 16 rows x 32 cols = two 16x16 WMMA
// tiles.  V_WMMA_F32_16X16X4_F32 computes D[M][N] = thr*(area_row[M] +
// area_col[N]) as a K=2 outer product (A: k0=thr*a_i, k1=1; B: k0=1,
// k1=thr*a_j; K=2..3 zero) -- co-executes with the VALU min/max intersection.
// D layout (lane l<16: M=v,N=l; lanes 16-31: M=v+8,N=l-16) lets a 32-bit
// __ballot per D-VGPR yield two 16-bit half-rows of the mask word.
// iou > thr  <=>  inter*(1+thr) > thr*(a_i + a_j).
// Class-offset trick == label-equality test (cross-class IoU forced 0).
// ---------------------------------------------------------------------------
__global__ __launch_bounds__(256) void k_mask(const float4* __restrict__ sbox,
                                              const int* __restrict__ slab,
                                              unsigned* __restrict__ mask) {
  __shared__ float4 srow[8][16];
  __shared__ int    srlab[8][16];
  const int lane = threadIdx.x & 31, wv = threadIdx.x >> 5;
  const int cw = blockIdx.x;
  const int rb = blockIdx.y * 8 + wv;
  const int b  = blockIdx.z;
  if (rb >= RB) return;
  if (cw * 32 + 31 <= rb * 16) return;           // strictly-lower triangle: no j>i
  const size_t bo = (size_t)b * NPTS;

  // Async-stage the 16 row boxes + labels this wave will broadcast-read.
  if (lane < 16) {
    unsigned lds = (unsigned)(unsigned long long)(uintptr_t)&srow[wv][lane];
    unsigned off = (unsigned)((bo + rb * 16 + lane) * sizeof(float4));
    async_lds_b128(lds, off, (unsigned long long)(uintptr_t)sbox);
  } else {
    unsigned lds = (unsigned)(unsigned long long)(uintptr_t)&srlab[wv][lane - 16];
    unsigned off = (unsigned)((bo + rb * 16 + (lane - 16)) * sizeof(int));
    async_lds_b32(lds, off, (unsigned long long)(uintptr_t)slab);
  }

  // Column boxes straight to VGPRs (overlaps with the async LDS fill; 537KB
  // of sorted boxes lives in the 192MB L2, so these are L2 hits).
  int jL = cw * 32 + (lane & 15), jR = jL + 16;
  int jLc = min(jL, NPTS - 1), jRc = min(jR, NPTS - 1);
  float4 cL = sbox[bo + jLc], cR = sbox[bo + jRc];
  int   clL = slab[bo + jLc], clR = slab[bo + jRc];
  float acL = (cL.z - cL.x) * (cL.w - cL.y);
  float acR = (cR.z - cR.x) * (cR.w - cR.y);

  wait_asynccnt0();

  const bool lo = lane < 16;
  float ar = 0.f;
  if (lo) { float4 rv = srow[wv][lane]; ar = (rv.z - rv.x) * (rv.w - rv.y); }

  // A (16x4 f32: lanes0-15 hold K0,K1; lanes16-31 hold K2,K3 = zero).
  v2f A  = { lo ? THR * ar  : 0.f, lo ? 1.f : 0.f };
  // B (4x16 f32, mirrored layout): K0 row = ones, K1 row = thr*area_col.
  v2f BL = { lo ? 1.f : 0.f, lo ? THR * acL : 0.f };
  v2f BR = { lo ? 1.f : 0.f, lo ? THR * acR : 0.f };
  v8f cz = {};
  v8f dl = __builtin_amdgcn_wmma_f32_16x16x4_f32(false, A, false, BL, (short)0, cz, false, false);
  v8f dr = __builtin_amdgcn_wmma_f32_16x16x4_f32(false, A, false, BR, (short)0, cz, false, false);

  const int hi8 = (lane >= 16) ? 8 : 0;
#pragma unroll
  for (int v = 0; v < 8; ++v) {
    float4 rv = srow[wv][v + hi8];
    int    rl = srlab[wv][v + hi8];
    float ixL = fminf(rv.z, cL.z) - fmaxf(rv.x, cL.x);
    float iyL = fminf(rv.w, cL.w) - fmaxf(rv.y, cL.y);
    float inL = fmaxf(ixL, 0.f) * fmaxf(iyL, 0.f);
    float ixR = fminf(rv.z, cR.z) - fmaxf(rv.x, cR.x);
    float iyR = fminf(rv.w, cR.w) - fmaxf(rv.y, cR.y);
    float inR = fmaxf(ixR, 0.f) * fmaxf(iyR, 0.f);
    bool pL = (inL * (1.f + THR) > dl[v]) && (rl == clL) && (jL < NPTS);
    bool pR = (inR * (1.f + THR) > dr[v]) && (rl == clR) && (jR < NPTS);
    unsigned mL = (unsigned)__ballot(pL);   // wave32: low 32 bits
    unsigned mR = (unsigned)__ballot(pR);
    if (lane == 0) {
      unsigned w0 = (mL & 0xFFFFu) | ((mR & 0xFFFFu) << 16);   // row rb*16+v
      unsigned w1 = (mL >> 16)     | (mR & 0xFFFF0000u);       // row rb*16+v+8
      int r0 = rb * 16 + v;
      mask[(bo + r0)     * NCW + cw] = w0;
      mask[(bo + r0 + 8) * NCW + cw] = w1;
    }
  }
}

// ---------------------------------------------------------------------------
// K4: greedy serial scan over the bitmask; one wave32 per image, running
// "removed" bitset in LDS.  Diagonal word masked to enforce j > i, so the
// j<=i bits K3 computed (incl. self-IoU=1) never suppress.
// ---------------------------------------------------------------------------
__global__ __launch_bounds__(32) void k_scan(const float* __restrict__ sskey,
                                             const unsigned* __restrict__ mask,
                                             unsigned* __restrict__ keep) {
  __shared__ unsigned removed[NCW];
  const int b = blockIdx.x, lane = threadIdx.x;
  const size_t bo = (size_t)b * NPTS;
  for (int i = lane; i < NCW; i += 32) removed[i] = 0u;
  __syncthreads();
  for (int i = 0; i < NPTS; ++i) {
    int w0 = i >> 5, bq = i & 31;
    bool alive = (sskey[bo + i] >= STHR) && (((removed[w0] >> bq) & 1u) == 0u);
    if (alive) {
      const unsigned* mrow = mask + (bo + i) * NCW;
      for (int ww = w0 + lane; ww < NCW; ww += 32) {
        unsigned m = mrow[ww];
        if (ww == w0) m &= (bq == 31) ? 0u : (0xFFFFFFFFu << (bq + 1)); // j > i only
        removed[ww] |= m;            // unique word per lane this iteration
      }
    }
    if (lane == 0) keep[bo + i] = alive ? 1u : 0u;
    __syncthreads();                 // 1-wave barrier: cheap
  }
}

// ---------------------------------------------------------------------------
// K5: pack outputs: dets[B][P][5] * keep, labels[B][P] (as f32), keep[B][P].
// ---------------------------------------------------------------------------
__global__ void k_out(const float* __restrict__ sskey, const int* __restrict__ slab,
                      const float4* __restrict__ sbox, const unsigned* __restrict__ keep,
                      float* __restrict__ out) {
  int t = blockIdx.x * blockDim.x + threadIdx.x;
  if (t >= BATCH * NPTS) return;
  float k = keep[t] ? 1.f : 0.f;
  float4 bx = sbox[t];
  float* d = out + (size_t)t * 5;
  d[0] = bx.x * k; d[1] = bx.y * k; d[2] = bx.z * k; d[3] = bx.w * k; d[4] = sskey[t] * k;
  out[(size_t)BATCH * NPTS * 5 + t] = (float)slab[t];
  out[(size_t)BATCH * NPTS * 6 + t] = k;
}

// ---------------------------------------------------------------------------
extern "C" void kernel_launch(void* const* d_in, const int* in_sizes, int n_in,
                              void* d_out, int out_size, void* d_ws, size_t ws_size,
                              hipStream_t stream) {
  (void)in_sizes; (void)n_in; (void)out_size; (void)ws_size;
  const float* c0 = (const float*)d_in[0];
  const float* c1 = (const float*)d_in[1];
  const float* c2 = (const float*)d_in[2];
  const float* r0 = (const float*)d_in[3];
  const float* r1 = (const float*)d_in[4];
  const float* r2 = (const float*)d_in[5];
  const float* o0 = (const float*)d_in[6];
  const float* o1 = (const float*)d_in[7];
  const float* o2 = (const float*)d_in[8];

  const size_t BP = (size_t)BATCH * NPTS;
  char* w = (char*)d_ws;                        // all blocks 16B-aligned
  float*    skey  = (float*)w;      w += BP * sizeof(float);
  int*      lab   = (int*)w;        w += BP * sizeof(int);
  float4*   box   = (float4*)w;     w += BP * sizeof(float4);
  float*    sskey = (float*)w;      w += BP * sizeof(float);
  int*      slab  = (int*)w;        w += BP * sizeof(int);
  float4*   sbox  = (float4*)w;     w += BP * sizeof(float4);
  unsigned* keep  = (unsigned*)w;   w += BP * sizeof(unsigned);
  unsigned* mask  = (unsigned*)w;   // BP * NCW u32 = 35.3 MB

  float* out = (float*)d_out;

  int nthr = BATCH * NPTS;
  k_decode<<<(nthr + 255) / 256, 256, 0, stream>>>(c0, c1, c2, r0, r1, r2, o0, o1, o2,
                                                   skey, lab, box);
  k_sort<<<BATCH, 512, 0, stream>>>(skey, lab, box, sskey, slab, sbox);
  dim3 g3(NCW, (RB + 7) / 8, BATCH);
  k_mask<<<g3, 256, 0, stream>>>(sbox, slab, mask);
  k_scan<<<BATCH, 32, 0, stream>>>(sskey, mask, keep);
  k_out<<<(nthr + 255) / 256, 256, 0, stream>>>(sskey, slab, sbox, keep, out);
}